// CLAHE_35390530519669
// MI455X (gfx1250) — compile-verified
//
#include <hip/hip_runtime.h>
#include <math.h>

// ---------------- problem constants (match reference) ----------------
#define IMG_H   2048
#define IMG_W   2048
#define GH      4
#define GW      4
#define TILE_H  512
#define TILE_W  512
#define NBINS   256
#define TILE_PIX (TILE_H * TILE_W)        // 262144
#define CLIP_MAX 1536u                    // int(1.5 * 262144 // 256)
#define TWO_PI_F 6.28318530717958647692f  // rounds to fp32(2*pi)

// Detect gfx1250 async global->LDS path (ASYNCcnt-tracked).
#if defined(__has_builtin)
#  if __has_builtin(__builtin_amdgcn_global_load_async_to_lds_b128)
#    define USE_ASYNC_LDS 1
#  endif
#endif

typedef int v4i __attribute__((vector_size(16)));        // 16-byte payload for b128
typedef __attribute__((address_space(1))) v4i* gv4i_p;   // global
typedef __attribute__((address_space(3))) v4i* lv4i_p;   // LDS

__device__ __forceinline__ void wait_asynccnt0() {
#if defined(__has_builtin) && __has_builtin(__builtin_amdgcn_s_wait_asynccnt)
  __builtin_amdgcn_s_wait_asynccnt(0);
#else
  asm volatile("s_wait_asynccnt 0" ::: "memory");
#endif
}

// jnp.remainder semantics: fmod, then add divisor on sign mismatch.
__device__ __forceinline__ float pymod(float x, float y) {
  float m = fmodf(x, y);
  if (m != 0.0f && ((m < 0.0f) != (y < 0.0f))) m += y;
  return m;
}

// ---------------- pass 0: zero the global histogram ----------------
__global__ __launch_bounds__(256) void clahe_zero(unsigned int* __restrict__ ghist) {
  int i = blockIdx.x * 256 + threadIdx.x;
  if (i < GH * GW * NBINS) ghist[i] = 0u;
}

// ---------------- pass 1: per-tile histogram of V = max(r,g,b)/255 ----------------
// Block = 256 threads, covers an 8-row x 512-col strip inside ONE tile.
// Grid = (4 tile-columns, 2048/8 row-strips).
__global__ __launch_bounds__(256) void clahe_hist(const float* __restrict__ img,
                                                  unsigned int* __restrict__ ghist) {
  __shared__ unsigned int lh[NBINS];
  const int tid = threadIdx.x;
  lh[tid] = 0u;                       // blockDim.x == NBINS == 256
  __syncthreads();

  const int tx      = blockIdx.x;          // 0..3 tile column
  const int rowBase = blockIdx.y * 8;      // first row of the strip
  const int tileId  = (rowBase / TILE_H) * GW + tx;

  for (int i = tid; i < 8 * TILE_W; i += 256) {
    const int r = i >> 9;            // /512
    const int c = i & 511;
    const int y = rowBase + r;
    const int x = tx * TILE_W + c;
    const size_t p = ((size_t)y * IMG_W + x) * 3;
    // Reference normalizes first, then takes max; fp32 divide is monotone.
    const float rn = img[p + 0] / 255.0f;
    const float gn = img[p + 1] / 255.0f;
    const float bn = img[p + 2] / 255.0f;
    const float v  = fmaxf(rn, fmaxf(gn, bn));
    const int bin  = (int)fminf(fmaxf(floorf(v * 256.0f), 0.0f), 255.0f);
    atomicAdd(&lh[bin], 1u);         // ds_add_u32
  }
  __syncthreads();
  atomicAdd(&ghist[tileId * NBINS + tid], lh[tid]);   // merge: 256 global atomics / block
}

// ---------------- pass 2: clip + redistribute + cumsum -> LUT ----------------
// One block (256 threads) per tile. All count math is integer-exact (< 2^24),
// matching the reference's fp32 arithmetic bit-for-bit at the final scale step.
// cumsum(hist_r)[i] = cumsum(min(h,1536))[i] + (i+1)*redist + min(i+1, residual)
__global__ __launch_bounds__(256) void clahe_lut(const unsigned int* __restrict__ ghist,
                                                 float* __restrict__ lut) {
  __shared__ unsigned int s[NBINS];
  const int t = blockIdx.x;
  const int i = threadIdx.x;
  const unsigned int hc = min(ghist[t * NBINS + i], CLIP_MAX);
  s[i] = hc;
  __syncthreads();
  // Hillis-Steele inclusive scan over 256 bins
  for (int off = 1; off < NBINS; off <<= 1) {
    unsigned int add = (i >= off) ? s[i - off] : 0u;
    __syncthreads();
    s[i] += add;
    __syncthreads();
  }
  const unsigned int total    = s[NBINS - 1];
  const unsigned int excess   = (unsigned int)TILE_PIX - total;
  const unsigned int redist   = excess >> 8;      // floor(excess/256)
  const unsigned int residual = excess & 255u;
  const unsigned int cum = s[i] + (unsigned int)(i + 1) * redist
                         + min((unsigned int)(i + 1), residual);
  const float lut_scale = 255.0f / 262144.0f;     // exact in fp32
  float l = floorf((float)cum * lut_scale);
  l = fminf(fmaxf(l, 0.0f), 255.0f);
  lut[t * NBINS + i] = l;
}

// ---------------- pass 3: RGB->HSV, bilinear LUT apply, HSV->RGB ----------------
// One block (256 threads) per image row. LUT (16 KB) staged into LDS via the
// gfx1250 async global->LDS path when available.
__global__ __launch_bounds__(256) void clahe_apply(const float* __restrict__ img,
                                                   const float* __restrict__ lut,
                                                   float* __restrict__ out) {
  // 16-byte aligned: b128 LDS writes (async copy) need natural alignment.
  __shared__ __align__(16) float slut[GH * GW * NBINS];   // 4096 floats = 16 KB
  const int tid = threadIdx.x;

#ifdef USE_ASYNC_LDS
  #pragma unroll
  for (int it = 0; it < 4; ++it) {
    const int idx = (it * 256 + tid) * 4;   // 16 bytes per lane per iter
    __builtin_amdgcn_global_load_async_to_lds_b128(
        (gv4i_p)(lut + idx),
        (lv4i_p)(slut + idx),
        0, 0);
  }
  wait_asynccnt0();
  __syncthreads();
#else
  for (int idx = tid; idx < GH * GW * NBINS; idx += 256) slut[idx] = lut[idx];
  __syncthreads();
#endif

  const int y = blockIdx.x;
  const float fy = ((float)y + 0.5f) / 512.0f - 0.5f;
  const float y0 = floorf(fy);
  const float wy = fy - y0;
  const int y0c = (int)fminf(fmaxf(y0,        0.0f), 3.0f);
  const int y1c = (int)fminf(fmaxf(y0 + 1.0f, 0.0f), 3.0f);

  for (int x = tid; x < IMG_W; x += 256) {
    const size_t p = ((size_t)y * IMG_W + x) * 3;
    const float rn = img[p + 0] / 255.0f;
    const float gn = img[p + 1] / 255.0f;
    const float bn = img[p + 2] / 255.0f;

    // ---- rgb_to_hsv ----
    const float maxc  = fmaxf(rn, fmaxf(gn, bn));
    const float minc  = fminf(rn, fminf(gn, bn));
    const float delta = maxc - minc;
    const float v     = maxc;
    const float sat   = delta / (maxc + 1e-8f);
    const float dsafe = (delta == 0.0f) ? 1.0f : delta;
    const float rc = maxc - rn, gc = maxc - gn, bc = maxc - bn;
    // first-occurrence argmax on the normalized channels (matches jnp.argmax)
    const float sel = (rn == maxc) ? (bc - gc)
                    : (gn == maxc) ? (rc - bc + 2.0f * dsafe)
                                   : (gc - rc + 4.0f * dsafe);
    const float h    = pymod(sel / dsafe, 6.0f);
    const float hrad = TWO_PI_F * (h / 6.0f);

    // ---- bilinear CLAHE LUT on V ----
    const int bin = (int)fminf(fmaxf(floorf(v * 256.0f), 0.0f), 255.0f);
    const float fx = ((float)x + 0.5f) / 512.0f - 0.5f;
    const float x0 = floorf(fx);
    const float wx = fx - x0;
    const int x0c = (int)fminf(fmaxf(x0,        0.0f), 3.0f);
    const int x1c = (int)fminf(fmaxf(x0 + 1.0f, 0.0f), 3.0f);
    const float v00 = slut[(y0c * GW + x0c) * NBINS + bin];
    const float v01 = slut[(y0c * GW + x1c) * NBINS + bin];
    const float v10 = slut[(y1c * GW + x0c) * NBINS + bin];
    const float v11 = slut[(y1c * GW + x1c) * NBINS + bin];
    const float vint = (1.0f - wy) * ((1.0f - wx) * v00 + wx * v01)
                     +        wy  * ((1.0f - wx) * v10 + wx * v11);
    const float veq = vint / 255.0f;

    // ---- hsv_to_rgb ----
    const float hh  = hrad / TWO_PI_F;
    const float h6  = hh * 6.0f;
    const float hif = pymod(floorf(h6), 6.0f);
    const float f   = pymod(h6, 6.0f) - hif;
    const int   hi  = (int)hif;
    const float pv = veq * (1.0f - sat);
    const float qv = veq * (1.0f - f * sat);
    const float tv = veq * (1.0f - (1.0f - f) * sat);
    float ro, go, bo;
    switch (hi) {
      case 0:  ro = veq; go = tv;  bo = pv;  break;
      case 1:  ro = qv;  go = veq; bo = pv;  break;
      case 2:  ro = pv;  go = veq; bo = tv;  break;
      case 3:  ro = pv;  go = qv;  bo = veq; break;
      case 4:  ro = tv;  go = pv;  bo = veq; break;
      default: ro = veq; go = pv;  bo = qv;  break;
    }
    out[p + 0] = ro;
    out[p + 1] = go;
    out[p + 2] = bo;
  }
}

// ---------------- launcher ----------------
extern "C" void kernel_launch(void* const* d_in, const int* in_sizes, int n_in,
                              void* d_out, int out_size, void* d_ws, size_t ws_size,
                              hipStream_t stream) {
  const float* img = (const float*)d_in[0];    // (2048, 2048, 3) fp32
  float* out = (float*)d_out;                  // (2048, 2048, 3) fp32

  // workspace: [0,16KB) uint hist[16][256]; [16KB,32KB) float lut[16][256]
  unsigned int* ghist = (unsigned int*)d_ws;
  float* lut = (float*)((char*)d_ws + (size_t)(GH * GW * NBINS) * sizeof(unsigned int));

  clahe_zero<<<(GH * GW * NBINS + 255) / 256, 256, 0, stream>>>(ghist);
  dim3 g1(GW, IMG_H / 8);
  clahe_hist<<<g1, 256, 0, stream>>>(img, ghist);
  clahe_lut<<<GH * GW, 256, 0, stream>>>(ghist, lut);
  clahe_apply<<<IMG_H, 256, 0, stream>>>(img, lut, out);
}